// AttentionConv2d_20624432955610
// MI455X (gfx1250) — compile-verified
//
#include <hip/hip_runtime.h>
#include <math.h>

typedef __attribute__((ext_vector_type(2))) float v2f;
typedef __attribute__((ext_vector_type(8))) float v8f;
typedef __attribute__((ext_vector_type(4))) unsigned int v4u;
typedef __attribute__((ext_vector_type(8))) int v8i;
typedef __attribute__((ext_vector_type(4))) int v4i;

// Problem constants
#define N_BATCH 16
#define D_IN    256
#define H_IMG   128
#define W_IMG   128
#define L_TOK   (H_IMG * W_IMG)   // 16384
#define D_OUT   256
#define NCHUNK  64                 // l-chunks of 256 per batch image
#define PITCH   258                // LDS row pitch (dwords) for conflict-free b64 frags

// Workspace layout (floats)
#define OFF_QK    0
#define OFF_QW2   256
#define OFF_HDOT  512
#define OFF_MZ    (512 + L_TOK)                   // 16896: [1024][2] (m, Z)
#define OFF_TPART (OFF_MZ + 2 * N_BATCH * NCHUNK) // 18944: [1024][256]

// Branch-free tanh-gelu: 0.5*a*(1+tanh(z)) == a * sigmoid(2z)
__device__ __forceinline__ float gelu_fast(float a) {
  float z = 0.79788456f * fmaf(0.044715f * a, a * a, a);
  return a * __builtin_amdgcn_rcpf(1.0f + __expf(-2.0f * z));
}

// XOR-butterfly add via ds_swizzle_b32 (group-of-32: offset = xor<<10 | and 0x1f)
template <int XOR>
__device__ __forceinline__ float swizzle_add(float v) {
  int iv = __builtin_bit_cast(int, v);
  int sv = __builtin_amdgcn_ds_swizzle(iv, (XOR << 10) | 0x1f);
  return v + __builtin_bit_cast(float, sv);
}

// ---------------------------------------------------------------------------
// Kernel 1: fold query into k_w and w2:  qk = k_w^T q,  qw2 = w2^T q
// ---------------------------------------------------------------------------
__global__ __launch_bounds__(256) void prep_kernel(const float* __restrict__ q,
                                                   const float* __restrict__ k_w,
                                                   const float* __restrict__ w2,
                                                   float* __restrict__ ws) {
  __shared__ float qs[256];
  int tid = threadIdx.x;
  qs[tid] = q[tid];
  __syncthreads();
  float a = 0.f, b = 0.f;
  for (int o = 0; o < 256; ++o) {
    float qo = qs[o];
    a = fmaf(qo, k_w[o * 256 + tid], a);
    b = fmaf(qo, w2[o * 256 + tid], b);
  }
  ws[OFF_QK + tid] = a;
  ws[OFF_QW2 + tid] = b;
}

// ---------------------------------------------------------------------------
// Kernel 2: Fourier positional MLP projected onto query.
// hdot[l] = gelu(feats[l] @ w1^T + b1) . qw2
// One workgroup = 16 l-rows; 16 waves, each owns one 16-col o-tile.
// K=256 accumulated with 64 x v_wmma_f32_16x16x4_f32 per wave.
// ---------------------------------------------------------------------------
__global__ __launch_bounds__(512, 1) void pos_kernel(const float* __restrict__ Wr,
                                                     const float* __restrict__ w1g,
                                                     const float* __restrict__ b1,
                                                     const float* __restrict__ qw2,
                                                     float* __restrict__ hdot) {
  __shared__ float w1s[256 * PITCH];    // 264 KB, padded rows
  __shared__ float feats[16 * PITCH];   // 16.5 KB
  __shared__ float wpart[16 * 16];      // per-(wave, m) partials

  int tid = threadIdx.x;
  int ltile = blockIdx.x * 16;

  // Stage w1 (row-major [o][j]) into padded LDS, coalesced float4.
  for (int k = 0; k < 32; ++k) {
    int idx = (k * 512 + tid) * 4;  // 65536 dwords total
    int o = idx >> 8, j = idx & 255;
    float4 v = *(const float4*)(w1g + idx);
    float* dst = &w1s[o * PITCH + j];
    dst[0] = v.x; dst[1] = v.y; dst[2] = v.z; dst[3] = v.w;
  }

  // Fourier features for 16 rows: feats[m][j], j<128 -> cos, else sin, scaled 1/16.
  for (int k = 0; k < 8; ++k) {
    int idx = k * 512 + tid;  // 0..4095
    int m = idx >> 8, j = idx & 255;
    int l = ltile + m;
    float gy = fmaf((float)(l >> 7), 2.0f / 127.0f, -1.0f);
    float gx = fmaf((float)(l & 127), 2.0f / 127.0f, -1.0f);
    int f = j & 127;
    float pr = gy * Wr[2 * f] + gx * Wr[2 * f + 1];
    float v = (j < 128) ? __cosf(pr) : __sinf(pr);
    feats[m * PITCH + j] = v * 0.0625f;
  }
  __syncthreads();

  int lane = tid & 31, wv = tid >> 5;
  int ln = lane & 15, hi = lane >> 4;

  // WMMA fragment bases (A: 16x4 f32, B: 4x16 f32, per-lane K = 2*hi + {0,1})
  const float* arow = &feats[ln * PITCH + 2 * hi];
  const float* brow = &w1s[(wv * 16 + ln) * PITCH + 2 * hi];

  v8f acc = {0.f, 0.f, 0.f, 0.f, 0.f, 0.f, 0.f, 0.f};
  #pragma unroll 4
  for (int k0 = 0; k0 < 256; k0 += 4) {
    v2f a; a.x = arow[k0]; a.y = arow[k0 + 1];
    v2f b; b.x = brow[k0]; b.y = brow[k0 + 1];
    acc = __builtin_amdgcn_wmma_f32_16x16x4_f32(false, a, false, b, (short)0, acc,
                                                false, false);
  }

  // Lane holds column o = wv*16 + ln; acc[r] is row m = r + 8*hi.
  int o = wv * 16 + ln;
  float b1o = b1[o];
  float qo = qw2[o];
  #pragma unroll
  for (int r = 0; r < 8; ++r) {
    float h = gelu_fast(acc[r] + b1o);
    float val = h * qo;
    // deterministic sum over the 16 lanes sharing row m (halves independent)
    val = swizzle_add<1>(val);
    val = swizzle_add<2>(val);
    val = swizzle_add<4>(val);
    val = swizzle_add<8>(val);
    if (ln == 0) wpart[wv * 16 + (r + 8 * hi)] = val;
  }
  __syncthreads();
  if (tid < 16) {
    float s = 0.f;
    for (int w = 0; w < 16; ++w) s += wpart[w * 16 + tid];
    hdot[ltile + tid] = s;
  }
}

// ---------------------------------------------------------------------------
// TDM helper: 2D tile load Global->LDS (rows of `tile_d0` contiguous elements,
// `tile_d1` rows of stride `row_stride` elements, 4-byte data).
// Descriptor packing per CDNA5 ISA section 8 (D# groups).
// ---------------------------------------------------------------------------
__device__ __forceinline__ void tdm_load_2d_f32(unsigned lds_byte_addr,
                                                const float* gptr,
                                                unsigned tensor_d0,
                                                unsigned tensor_d1,
                                                unsigned row_stride,
                                                unsigned tile_d0,
                                                unsigned tile_d1) {
  unsigned long long ga = (unsigned long long)(size_t)gptr;
  v4u g0;
  g0.x = 1u;                                            // count=1, user mode
  g0.y = lds_byte_addr;                                 // lds_addr
  g0.z = (unsigned)(ga & 0xFFFFFFFFu);                  // global_addr[31:0]
  g0.w = (unsigned)((ga >> 32) & 0x1FFFFFFu) | (2u << 30); // addr[56:32] | type=2
  v8i g1;
  g1[0] = (int)(2u << 16);                              // data_size=4B, mask=0
  g1[1] = (int)((tensor_d0 & 0xFFFFu) << 16);           // tensor_dim0[15:0]
  g1[2] = (int)((tensor_d0 >> 16) | ((tensor_d1 & 0xFFFFu) << 16));
  g1[3] = (int)((tensor_d1 >> 16) | (tile_d0 << 16));   // tile_dim0
  g1[4] = (int)(tile_d1);                               // tile_dim1 (tile_dim2=0)
  g1[5] = (int)(row_stride);                            // tensor_dim0_stride[31:0]
  g1[6] = 0;                                            // stride0[47:32]|stride1 lo
  g1[7] = 0;
  v4i gz4 = {0, 0, 0, 0};
  v8i gz8 = {0, 0, 0, 0, 0, 0, 0, 0};
  __builtin_amdgcn_tensor_load_to_lds(g0, g1, gz4, gz4, gz8, 0);
}

// ---------------------------------------------------------------------------
// Kernel 3: single-pass streaming attention over x with online softmax.
// Block = (n, chunk of 256 l). TDM streams the 256x256 f32 tile into LDS
// (4 descriptors, one per wave 0-3), then scores + local softmax stats +
// weighted token partial t_b[d] are computed from LDS.
// ---------------------------------------------------------------------------
__global__ __launch_bounds__(256, 1) void fuse_kernel(const float* __restrict__ x,
                                                      const float* __restrict__ qk,
                                                      const float* __restrict__ hdot,
                                                      float* __restrict__ tpart,
                                                      float* __restrict__ mz) {
  __shared__ float tile[256 * 256];  // 256 KB: tile[d][l]
  __shared__ float qk_s[256];
  __shared__ float p_s[256];
  __shared__ float red[256];

  int tid = threadIdx.x;
  int wv = tid >> 5;
  int n = blockIdx.x >> 6, c = blockIdx.x & 63;
  int l0 = c << 8;

  qk_s[tid] = qk[tid];

  // Async-tensor load: waves 0..3 each DMA a 64-row sub-tile into LDS.
  if (wv < 4) {
    const float* gsub = x + (((size_t)n * 256 + (size_t)(wv * 64)) << 14) + l0;
    unsigned lds_addr = (unsigned)(size_t)&tile[wv * 64 * 256];
    tdm_load_2d_f32(lds_addr, gsub,
                    /*tensor_d0=*/L_TOK, /*tensor_d1=*/64,
                    /*row_stride=*/L_TOK, /*tile_d0=*/256, /*tile_d1=*/64);
    __builtin_amdgcn_s_wait_tensorcnt(0);
  }
  __syncthreads();

  // Phase 1: score for l = l0 + tid (column read, conflict-free).
  float s = 0.f;
  #pragma unroll 8
  for (int d = 0; d < 256; ++d) s = fmaf(qk_s[d], tile[(d << 8) + tid], s);
  s = (s + hdot[l0 + tid]) * 0.0625f;

  // Phase 2: local max and sum of exp (deterministic tree reductions).
  red[tid] = s;
  __syncthreads();
  for (int st = 128; st > 0; st >>= 1) {
    if (tid < st) red[tid] = fmaxf(red[tid], red[tid + st]);
    __syncthreads();
  }
  float m = red[0];
  __syncthreads();
  float e = __expf(s - m);
  p_s[tid] = e;
  red[tid] = e;
  __syncthreads();
  for (int st = 128; st > 0; st >>= 1) {
    if (tid < st) red[tid] += red[tid + st];
    __syncthreads();
  }
  float Z = red[0];

  // Phase 3: t_b[d=tid] = sum_l p[l] * tile[d][l]; rotated l avoids bank conflicts.
  float acc = 0.f;
  #pragma unroll 8
  for (int i = 0; i < 256; ++i) {
    int l = (i + tid) & 255;
    acc = fmaf(p_s[l], tile[(tid << 8) + l], acc);
  }
  int b = (n << 6) + c;
  tpart[(b << 8) + tid] = acc;
  if (tid == 0) { mz[2 * b] = m; mz[2 * b + 1] = Z; }
}

// ---------------------------------------------------------------------------
// Kernel 4: merge chunk partials per n (log-sum-exp) and apply v_w, v_b.
// ---------------------------------------------------------------------------
__global__ __launch_bounds__(256) void merge_kernel(const float* __restrict__ mz,
                                                    const float* __restrict__ tpart,
                                                    const float* __restrict__ v_w,
                                                    const float* __restrict__ v_b,
                                                    float* __restrict__ out) {
  __shared__ float mbuf[NCHUNK];
  __shared__ float zbuf[NCHUNK];
  __shared__ float wbuf[NCHUNK];
  __shared__ float tnorm[256];

  int tid = threadIdx.x;
  int n = blockIdx.x;
  if (tid < NCHUNK) {
    mbuf[tid] = mz[2 * (n * NCHUNK + tid)];
    zbuf[tid] = mz[2 * (n * NCHUNK + tid) + 1];
  }
  __syncthreads();
  float m = -1e30f;
  for (int c = 0; c < NCHUNK; ++c) m = fmaxf(m, mbuf[c]);
  float Z = 0.f;
  for (int c = 0; c < NCHUNK; ++c) Z += __expf(mbuf[c] - m) * zbuf[c];
  __syncthreads();
  if (tid < NCHUNK) wbuf[tid] = __expf(mbuf[tid] - m) / Z;
  __syncthreads();

  float acc = 0.f;
  for (int c = 0; c < NCHUNK; ++c)
    acc = fmaf(wbuf[c], tpart[((size_t)(n * NCHUNK + c) << 8) + tid], acc);
  tnorm[tid] = acc;
  __syncthreads();

  float r = v_b[tid];
  for (int d = 0; d < 256; ++d) r = fmaf(v_w[tid * 256 + d], tnorm[d], r);
  out[n * 256 + tid] = r;
}

// ---------------------------------------------------------------------------
extern "C" void kernel_launch(void* const* d_in, const int* in_sizes, int n_in,
                              void* d_out, int out_size, void* d_ws, size_t ws_size,
                              hipStream_t stream) {
  const float* x     = (const float*)d_in[0];
  const float* query = (const float*)d_in[1];
  const float* k_w   = (const float*)d_in[2];
  // d_in[3] = k_b : constant under softmax, unused
  const float* v_w   = (const float*)d_in[4];
  const float* v_b   = (const float*)d_in[5];
  const float* Wr    = (const float*)d_in[6];
  const float* w1    = (const float*)d_in[7];
  const float* b1    = (const float*)d_in[8];
  const float* w2    = (const float*)d_in[9];
  // d_in[10] = b2 : constant under softmax, unused
  float* out = (float*)d_out;
  float* ws  = (float*)d_ws;

  prep_kernel<<<1, 256, 0, stream>>>(query, k_w, w2, ws);
  pos_kernel<<<L_TOK / 16, 512, 0, stream>>>(Wr, w1, b1, ws + OFF_QW2, ws + OFF_HDOT);
  fuse_kernel<<<N_BATCH * NCHUNK, 256, 0, stream>>>(x, ws + OFF_QK, ws + OFF_HDOT,
                                                    ws + OFF_TPART, ws + OFF_MZ);
  merge_kernel<<<N_BATCH, 256, 0, stream>>>(ws + OFF_MZ, ws + OFF_TPART, v_w, v_b, out);
}